// ResidualFSQ_6700148982003
// MI455X (gfx1250) — compile-verified
//
#include <hip/hip_runtime.h>
#include <math.h>

// CDNA5 / gfx1250 fused Residual-FSQ.
//  x --(TDM tensor_load_to_lds, double-buffered, LDS-pad stride 68)--> LDS
//    --(v_wmma_f32_16x16x4_f32)--> z --(8x FSQ rounds, v_tanh_f32, xor-shuffle
//    index reduce)--> q --(WMMA)--> out + b_out.
// Memory-bound: 128MB in + 129MB out @ 23.3 TB/s => ~11us floor.

typedef __attribute__((ext_vector_type(2))) float v2f;
typedef __attribute__((ext_vector_type(8))) float v8f;
typedef __attribute__((ext_vector_type(4))) unsigned int u32x4;
typedef __attribute__((ext_vector_type(4))) int i32x4;
typedef __attribute__((ext_vector_type(8))) int i32x8;

#define DIM 1024
#define NQ 8
#define TOK_PER_BLOCK 128            // 8 waves x 16 tokens
#define CHUNK 64                     // K-columns per TDM chunk
#define NCHUNK (DIM / CHUNK)         // 16
#define XROWSTR (CHUNK + 4)          // TDM pads 4 dwords after every 64 dwords
#define XBUF_FLOATS (TOK_PER_BLOCK * XROWSTR)
#define WIN_STR 1032                 // W_in LDS row stride (rows 8 banks apart)
#define QSTR 10                      // q staging stride (8B aligned, no conflicts)
#define WOUT_ZOFF (DIM * 6)          // zeroed pair appended to lds_wout

// Hardware V_TANH_F32 (TRANS32, co-executes with WMMA) when the builtin exists.
#if defined(__has_builtin)
#  if __has_builtin(__builtin_amdgcn_tanhf)
#    define TANHF(x) __builtin_amdgcn_tanhf(x)
#  elif __has_builtin(__builtin_amdgcn_tanh_f32)
#    define TANHF(x) __builtin_amdgcn_tanh_f32(x)
#  else
#    define TANHF(x) tanhf(x)
#  endif
#else
#  define TANHF(x) tanhf(x)
#endif

// Issue one TDM load: 128 rows x 64 cols of f32 from a row-major [.,1024]
// tensor into LDS at lds_off, padded to 68-float rows. D# per ISA ch.8.
__device__ __forceinline__ void tdm_issue_chunk(const float* gsrc, unsigned lds_off)
{
    unsigned long long ga = (unsigned long long)(uintptr_t)gsrc;
    u32x4 g0;
    g0.x = 1u;                                                 // count=1 (user D#)
    g0.y = lds_off;                                            // LDS byte address
    g0.z = (unsigned)(ga & 0xffffffffu);                       // global_addr[31:0]
    g0.w = (unsigned)((ga >> 32) & 0x01ffffffu) | (2u << 30);  // [56:32] | type=2
    i32x8 g1;
    g1[0] = (2 << 16)        // data_size = 4B
          | (1 << 20)        // pad_enable
          | (5 << 22)        // pad_interval = 64 dwords
          | (3 << 25);       // pad_amount   = 4 dwords -> row stride 68 floats
    g1[1] = (int)((unsigned)CHUNK << 16);          // tensor_dim0 = 64
    g1[2] = (int)((unsigned)TOK_PER_BLOCK << 16);  // tensor_dim1 = 128
    g1[3] = (int)((unsigned)CHUNK << 16);          // tile_dim0   = 64
    g1[4] = TOK_PER_BLOCK;                         // tile_dim1   = 128
    g1[5] = DIM;                                   // tensor_dim0_stride = 1024
    g1[6] = 0;
    g1[7] = 0;
    i32x4 z4 = {0, 0, 0, 0};
#if defined(__clang_major__) && (__clang_major__ >= 23)
    i32x8 z8 = {0, 0, 0, 0, 0, 0, 0, 0};
    __builtin_amdgcn_tensor_load_to_lds(g0, g1, z4, z4, z8, 0);
#else
    __builtin_amdgcn_tensor_load_to_lds(g0, g1, z4, z4, 0);
#endif
}

__global__ __launch_bounds__(256) void rfsq_kernel(
    const float* __restrict__ x,
    const float* __restrict__ W_in,    // [6,1024]
    const float* __restrict__ b_in,    // [6]
    const float* __restrict__ W_out,   // [1024,6]
    const float* __restrict__ b_out,   // [1024]
    float* __restrict__ out,           // [tokens,1024]
    float* __restrict__ idx_out)       // [tokens,8] (ints stored as f32)
{
    __shared__ float xbuf[2 * XBUF_FLOATS];     // TDM double buffer (2 x 34KB)
    __shared__ float lds_win[7 * WIN_STR];      // rows 0..5 = W_in, row 6 = zeros
    __shared__ float lds_wout[DIM * 6 + 8];     // + zeroed tail for K{6,7}
    __shared__ float lds_bout[DIM];
    __shared__ float q_lds[TOK_PER_BLOCK * QSTR];

    const int tid = threadIdx.x;
    const int w    = tid >> 5;
    const int lane = tid & 31;
    const int hl   = lane >> 4;     // lane half (K split per ISA 7.12.2)
    const int n    = lane & 15;     // B/C/D column id; A row (token) id
    const int t0   = blockIdx.x * TOK_PER_BLOCK + w * 16;
    const float* xblock = x + (size_t)blockIdx.x * TOK_PER_BLOCK * DIM;

    // ---- kick off the TDM pipeline before staging weights ----
    if (w == 0) {
        tdm_issue_chunk(xblock + 0 * CHUNK, (unsigned)(uintptr_t)&xbuf[0]);
        tdm_issue_chunk(xblock + 1 * CHUNK, (unsigned)(uintptr_t)&xbuf[XBUF_FLOATS]);
    }

    // ---- stage weights/bias into LDS (coalesced) ----
    for (int i = tid; i < 6 * DIM; i += 256) {
        int r = i >> 10, c = i & (DIM - 1);
        lds_win[r * WIN_STR + c] = W_in[i];
    }
    for (int i = tid; i < DIM; i += 256) lds_win[6 * WIN_STR + i] = 0.f; // zero row
    for (int i = tid; i < DIM * 6; i += 256) lds_wout[i] = W_out[i];
    if (tid < 8) lds_wout[WOUT_ZOFF + tid] = 0.f;                       // zero tail
    for (int i = tid; i < DIM; i += 256) lds_bout[i] = b_out[i];

    // ---------------- project_in: z[16tok x 6] = x * W_in^T (WMMA) ---------
    // A: lane<16 -> M=n,K{0,1}; lane>=16 -> M=n,K{2,3}
    // B: VGPR j, lanes<16 row K=j, lanes>=16 row K=2+j; cols 6..15 read zero row
    const int wrow = (n < 6) ? n : 6;
    v8f acc = {};
    for (int ci = 0; ci < NCHUNK; ++ci) {
        if (w == 0) {
            if (ci < NCHUNK - 1) __builtin_amdgcn_s_wait_tensorcnt(1);
            else                 __builtin_amdgcn_s_wait_tensorcnt(0);
        }
        __syncthreads();   // chunk ci resident (and weights staged, ci==0)
        const float* xb = &xbuf[(ci & 1) * XBUF_FLOATS + (w * 16 + n) * XROWSTR + 2 * hl];
        const float* wb = &lds_win[wrow * WIN_STR + ci * CHUNK + 2 * hl];
        #pragma unroll
        for (int k = 0; k < CHUNK; k += 4) {
            v2f a = *(const v2f*)(xb + k);
            v2f b = *(const v2f*)(wb + k);
            acc = __builtin_amdgcn_wmma_f32_16x16x4_f32(false, a, false, b,
                                                        (short)0, acc, false, false);
        }
        __syncthreads();   // everyone done with buf (ci&1) before it is refilled
        if (w == 0 && ci + 2 < NCHUNK)
            tdm_issue_chunk(xblock + (ci + 2) * CHUNK,
                            (unsigned)(uintptr_t)&xbuf[(ci & 1) * XBUF_FLOATS]);
    }

    const float bin = (n < 6) ? b_in[n] : 0.f;

    // ---------------- per-lane FSQ constants (code dim = n; 0..5 real) -----
    const float lvl    = (n < 3) ? 8.f : 5.f;
    const float hw     = (n < 3) ? 4.f : 2.f;
    const float half_l = (lvl - 1.f) * (1.f + 1e-3f) * 0.5f;
    const float off    = (n < 3) ? 0.5f : 0.f;
    const float shift  = atanhf(off / half_l);
    const float inv_hw = 1.f / hw;
    const float lm1    = lvl - 1.f;
    const float ilm1   = 1.f / lm1;
    const float basis  = (n == 0) ? 1.f : (n == 1) ? 8.f : (n == 2) ? 64.f :
                         (n == 3) ? 512.f : (n == 4) ? 2560.f :
                         (n == 5) ? 12800.f : 0.f;

    // D layout: VGPR r = token (hl*8 + r), lane = code dim n -> elementwise FSQ
    float res[8], qout[8], keep[8];
    #pragma unroll
    for (int r = 0; r < 8; ++r) { res[r] = acc[r] + bin; qout[r] = 0.f; keep[r] = 0.f; }

    float scale = 1.f, rscale = 1.f;
    for (int q = 0; q < NQ; ++q) {
        #pragma unroll
        for (int r = 0; r < 8; ++r) {
            float bounded = TANHF(res[r] * rscale + shift) * half_l - off;
            float rounded = rintf(bounded);              // round-half-even
            float code    = rounded * inv_hw;
            float quant   = code * scale;
            res[r]  -= quant;
            qout[r] += quant;
            // index contribution: zhat*basis = (rounded + hw)*basis; lanes 6,7
            // contribute 0 so a 3-step butterfly over lanes {0..7} suffices.
            float v = (n < 6) ? (rounded + hw) * basis : 0.f;
            v += __shfl_xor(v, 1);
            v += __shfl_xor(v, 2);
            v += __shfl_xor(v, 4);
            keep[r] = (n == q) ? v : keep[r];            // lane n holds q = n
        }
        scale  *= ilm1;       // (levels-1)^(-q)
        rscale *= lm1;        // (levels-1)^(+q)
    }

    // batched index stores: lane n (< 8) owns quantizer q = n
    if (n < NQ) {
        #pragma unroll
        for (int r = 0; r < 8; ++r)
            idx_out[(size_t)(t0 + hl * 8 + r) * NQ + n] = keep[r];
    }

    // ---------------- stage q for out-projection A fragments ---------------
    if (n < 8) {
        #pragma unroll
        for (int r = 0; r < 8; ++r) {
            float val = (n < 6) ? qout[r] : 0.f;          // pad K=6,7 with zeros
            q_lds[(w * 16 + hl * 8 + r) * QSTR + n] = val;
        }
    }
    __syncthreads();

    // ---------------- project_out: out = q * W_out^T + b_out (WMMA) --------
    const v2f a0 = *(const v2f*)&q_lds[(w * 16 + n) * QSTR + 2 * hl];      // K{0,1}/{2,3}
    const v2f a1 = *(const v2f*)&q_lds[(w * 16 + n) * QSTR + 4 + 2 * hl];  // K{4,5}/{6,7}

    #pragma unroll 2
    for (int dt = 0; dt < DIM / 16; ++dt) {
        const int d = dt * 16 + n;
        v2f b0 = *(const v2f*)&lds_wout[d * 6 + 2 * hl];
        const int off1 = hl ? WOUT_ZOFF : (d * 6 + 4);   // branchless K{6,7}=0
        v2f b1 = *(const v2f*)&lds_wout[off1];
        v8f o = {};
        o = __builtin_amdgcn_wmma_f32_16x16x4_f32(false, a0, false, b0,
                                                  (short)0, o, false, false);
        o = __builtin_amdgcn_wmma_f32_16x16x4_f32(false, a1, false, b1,
                                                  (short)0, o, false, false);
        const float bo = lds_bout[d];
        #pragma unroll
        for (int r = 0; r < 8; ++r)
            out[(size_t)(t0 + hl * 8 + r) * DIM + d] = o[r] + bo;
    }
}

extern "C" void kernel_launch(void* const* d_in, const int* in_sizes, int n_in,
                              void* d_out, int out_size, void* d_ws, size_t ws_size,
                              hipStream_t stream) {
    const float* x     = (const float*)d_in[0];
    const float* W_in  = (const float*)d_in[1];
    const float* b_in  = (const float*)d_in[2];
    const float* W_out = (const float*)d_in[3];
    const float* b_out = (const float*)d_in[4];
    float* out = (float*)d_out;

    const int n_tokens = in_sizes[0] / DIM;               // 32768
    float* idx_out = out + (size_t)n_tokens * DIM;        // tuple outputs concat
    const int blocks = n_tokens / TOK_PER_BLOCK;          // 256

    rfsq_kernel<<<blocks, 256, 0, stream>>>(x, W_in, b_in, W_out, b_out,
                                            out, idx_out);
    (void)n_in; (void)out_size; (void)d_ws; (void)ws_size;
}